// PromptGenerator_Encoder_27728308863452
// MI455X (gfx1250) — compile-verified
//
#include <hip/hip_runtime.h>
#include <stdint.h>

// ---- problem constants (match reference) ----
#define B_      2
#define C_      4
#define S_      (128*128*128)     // 2,097,152 voxels per (b,c) plane
#define NPTS    32
#define THRV    0.5f
#define HISTB   2048              // radix-select buckets over mantissa bits [22:12]
#define TPB     256               // 8 wave32 waves per block

// voxels per TDM tile: 4 channels x 1024 voxels x 4B = 16KB per tile
#define TILE_V  1024
#define ITERS   16                // tiles per block -> 16384 voxels/block
#define NCHUNK  (S_ / (ITERS * TILE_V))   // 128 blocks per batch

// ---- workspace layout (uint32 units) ----
#define WS_HIST   0
#define WS_THRESH (WS_HIST + B_*C_*HISTB)     // 16384
#define WS_CNT    (WS_THRESH + B_*C_)         // +8
#define WS_VALID  (WS_CNT + B_*C_)            // +8
#define WS_SEL    (WS_VALID + B_*C_)          // +8
#define WS_CAND   (WS_SEL + B_*C_*NPTS)       // +256 -> 16664

typedef uint32_t v4u __attribute__((ext_vector_type(4)));
typedef int32_t  v8i __attribute__((ext_vector_type(8)));
typedef __attribute__((address_space(3))) float lds_f32;

#if __has_builtin(__builtin_amdgcn_s_wait_tensorcnt)
#define WAIT_TENSORCNT(n) __builtin_amdgcn_s_wait_tensorcnt(n)
#else
#define WAIT_TENSORCNT(n) asm volatile("s_wait_tensorcnt %0" :: "i"(n) : "memory")
#endif

// Issue one TDM 2D tile load: 4 rows (channels, stride S_ elements) x 1024 f32.
// D# group0: count=1 | lds_addr | global_addr(57b) | type=2
// D# group1: data_size=2(4B), tensor_dim0=S_, tensor_dim1=C_, tile_dim0=1024,
//            tile_dim1=C_, tensor_dim0_stride=S_
__device__ __forceinline__ void tdm_issue_tile(uint64_t gaddr, uint32_t laddr)
{
    v4u g0;
    g0[0] = 1u;                                            // count=1, user mode
    g0[1] = laddr;                                         // LDS byte address
    g0[2] = (uint32_t)gaddr;                               // global_addr[31:0]
    g0[3] = ((uint32_t)(gaddr >> 32) & 0x01FFFFFFu)        // global_addr[56:32]
            | 0x80000000u;                                 // type=2 ("image")
    v8i g1;
    g1[0] = 0x00020000;                                    // data_size=2 (4B)
    g1[1] = 0;                                             // dim0[15:0]=0 (S_ low16)
    g1[2] = (int)(((uint32_t)(S_ >> 16)) | ((uint32_t)C_ << 16)); // dim0 hi | dim1 lo
    g1[3] = (int)(1024u << 16);                            // dim1 hi=0 | tile_dim0=1024
    g1[4] = (int)C_;                                       // tile_dim1=4, tile_dim2=0
    g1[5] = (int)S_;                                       // dim0_stride low32
    g1[6] = 0;                                             // dim0_stride hi | dim1_stride lo
    g1[7] = 0;
    asm volatile("tensor_load_to_lds %0, %1" :: "s"(g0), "s"(g1) : "memory");
}

__device__ __forceinline__ uint32_t lds_byte_offset(float* p)
{
    return (uint32_t)(size_t)(lds_f32*)p;   // addrspacecast generic->LDS, ptrtoint
}

__device__ __forceinline__ uint32_t prob_bucket(uint32_t keybits)
{
    uint32_t bk = (keybits - 0x3F000001u) >> 12;   // p in (0.5,1] -> [0,2047]
    return bk > (HISTB - 1u) ? (HISTB - 1u) : bk;
}

// softmax over the 4 channels of one voxel staged in LDS
__device__ __forceinline__ void softmax4(const float* st, int v, float p[4])
{
    float x0 = st[0 * TILE_V + v];
    float x1 = st[1 * TILE_V + v];
    float x2 = st[2 * TILE_V + v];
    float x3 = st[3 * TILE_V + v];
    float m  = fmaxf(fmaxf(x0, x1), fmaxf(x2, x3));
    float e0 = __expf(x0 - m), e1 = __expf(x1 - m);
    float e2 = __expf(x2 - m), e3 = __expf(x3 - m);
    float inv = 1.0f / (e0 + e1 + e2 + e3);
    p[0] = e0 * inv; p[1] = e1 * inv; p[2] = e2 * inv; p[3] = e3 * inv;
}

// ---------------- pass 1: TDM-streamed softmax + per-(b,c) histogram ----------
__global__ __launch_bounds__(TPB) void pass1_hist(const float* __restrict__ logits,
                                                  uint32_t* __restrict__ hist)
{
    __shared__ uint32_t lhist[C_ * HISTB];        // 32 KB
    __shared__ float    stage[2][C_ * TILE_V];    // 2 x 16 KB double buffer

    const int t = threadIdx.x;
    const int b = blockIdx.y;
    const int chunkBase = blockIdx.x * (ITERS * TILE_V);

    for (int i = t; i < C_ * HISTB; i += TPB) lhist[i] = 0;
    __syncthreads();

    const uint64_t gbase = (uint64_t)(size_t)(logits + (size_t)b * C_ * S_);
    const uint32_t lbase0 = lds_byte_offset(&stage[0][0]);
    const uint32_t lbase1 = lds_byte_offset(&stage[1][0]);

    if (t < 32)   // wave 0 drives the TDM pipeline (descriptor is wave-uniform)
        tdm_issue_tile(gbase + (uint64_t)chunkBase * 4u, lbase0);

    for (int i = 0; i < ITERS; ++i) {
        if (t < 32) {
            if (i + 1 < ITERS) {
                tdm_issue_tile(gbase + (uint64_t)(chunkBase + (i + 1) * TILE_V) * 4u,
                               ((i + 1) & 1) ? lbase1 : lbase0);
                WAIT_TENSORCNT(1);    // tile i complete (TENSORcnt is in-order)
            } else {
                WAIT_TENSORCNT(0);
            }
        }
        __syncthreads();              // publish tile i to all waves

        const float* st = &stage[i & 1][0];
        #pragma unroll
        for (int j = 0; j < 4; ++j) {
            const int v = t * 4 + j;                  // voxel within tile
            float p[4];
            softmax4(st, v, p);
            #pragma unroll
            for (int c = 0; c < 4; ++c) {
                if (p[c] > THRV) {
                    uint32_t bk = prob_bucket(__float_as_uint(p[c]));
                    atomicAdd(&lhist[c * HISTB + bk], 1u);
                }
            }
        }
        __syncthreads();              // all consumed before buffer is overwritten
    }

    uint32_t* gh = hist + (size_t)b * (C_ * HISTB);
    for (int i = t; i < C_ * HISTB; i += TPB) {
        uint32_t v = lhist[i];
        if (v) atomicAdd(&gh[i], v);
    }
}

// ---------------- pass 2: find threshold bucket per (b,c) --------------------
__global__ void select_thresh(const uint32_t* __restrict__ hist,
                              uint32_t* __restrict__ thresh)
{
    int i = threadIdx.x;
    if (i < B_ * C_) {
        const uint32_t* h = hist + (size_t)i * HISTB;
        uint32_t acc = 0, T = 0;
        for (int bk = HISTB - 1; bk >= 0; --bk) {
            acc += h[bk];
            if (acc >= NPTS) { T = (uint32_t)bk; break; }
        }
        thresh[i] = T;   // total < 32 -> T=0 -> collect everything
    }
}

// ---------------- pass 3: TDM re-stream, gather candidates >= threshold ------
__global__ __launch_bounds__(TPB) void collect(const float* __restrict__ logits,
                                               const uint32_t* __restrict__ thresh,
                                               uint32_t* __restrict__ cnt,
                                               uint32_t* __restrict__ candKey,
                                               uint32_t* __restrict__ candIdx,
                                               int cap)
{
    __shared__ float stage[2][C_ * TILE_V];       // 2 x 16 KB double buffer

    const int t = threadIdx.x;
    const int b = blockIdx.y;
    uint32_t th[4] = { thresh[b*4+0], thresh[b*4+1], thresh[b*4+2], thresh[b*4+3] };
    const int chunkBase = blockIdx.x * (ITERS * TILE_V);

    const uint64_t gbase = (uint64_t)(size_t)(logits + (size_t)b * C_ * S_);
    const uint32_t lbase0 = lds_byte_offset(&stage[0][0]);
    const uint32_t lbase1 = lds_byte_offset(&stage[1][0]);

    if (t < 32)
        tdm_issue_tile(gbase + (uint64_t)chunkBase * 4u, lbase0);

    for (int i = 0; i < ITERS; ++i) {
        if (t < 32) {
            if (i + 1 < ITERS) {
                tdm_issue_tile(gbase + (uint64_t)(chunkBase + (i + 1) * TILE_V) * 4u,
                               ((i + 1) & 1) ? lbase1 : lbase0);
                WAIT_TENSORCNT(1);
            } else {
                WAIT_TENSORCNT(0);
            }
        }
        __syncthreads();

        const float* st = &stage[i & 1][0];
        #pragma unroll
        for (int j = 0; j < 4; ++j) {
            const int v = t * 4 + j;
            const uint32_t vox = (uint32_t)(chunkBase + i * TILE_V + v);
            float p[4];
            softmax4(st, v, p);
            #pragma unroll
            for (int c = 0; c < 4; ++c) {
                if (p[c] > THRV) {
                    uint32_t key = __float_as_uint(p[c]);
                    uint32_t bk  = prob_bucket(key);
                    if (bk >= th[c]) {
                        uint32_t pos = atomicAdd(&cnt[b*4 + c], 1u);
                        if (pos < (uint32_t)cap) {
                            size_t o = (size_t)(b*4 + c) * cap + pos;
                            candKey[o] = key;
                            candIdx[o] = vox;
                        }
                    }
                }
            }
        }
        __syncthreads();
    }
}

// ---------------- pass 4: per (b, ordered-class) top-32 by repeated argmax ---
__global__ __launch_bounds__(TPB) void topk(const uint32_t* __restrict__ cnt,
                                            uint32_t* __restrict__ candKey,
                                            const uint32_t* __restrict__ candIdx,
                                            uint32_t* __restrict__ sel,
                                            uint32_t* __restrict__ valid,
                                            int cap)
{
    __shared__ unsigned long long sR[TPB];
    __shared__ int sP[TPB];
    const int t  = threadIdx.x;
    const int bc = blockIdx.x;          // b*4 + oc
    const int b  = bc >> 2, oc = bc & 3;
    const int c  = (oc + 1) & 3;        // visit order [1,2,3,0]
    uint32_t n32 = cnt[b*4 + c];
    int n = (n32 > (uint32_t)cap) ? cap : (int)n32;
    uint32_t* key = candKey + (size_t)(b*4 + c) * cap;
    const uint32_t* idx = candIdx + (size_t)(b*4 + c) * cap;
    const int K = n < NPTS ? n : NPTS;

    for (int k = 0; k < K; ++k) {
        unsigned long long best = 0; int bp = -1;
        for (int i = t; i < n; i += TPB) {
            uint32_t kv = key[i];
            if (kv) {
                // desc value, ties -> lower flat index first (jax top_k order)
                unsigned long long r = ((unsigned long long)kv << 32) | (uint32_t)(~idx[i]);
                if (r > best) { best = r; bp = i; }
            }
        }
        sR[t] = best; sP[t] = bp;
        __syncthreads();
        for (int off = TPB / 2; off > 0; off >>= 1) {
            if (t < off && sR[t + off] > sR[t]) { sR[t] = sR[t + off]; sP[t] = sP[t + off]; }
            __syncthreads();
        }
        if (t == 0) {
            sel[(b*4 + oc) * NPTS + k] = ~(uint32_t)sR[0];   // recover flat voxel idx
            key[sP[0]] = 0;                                  // remove winner
        }
        __syncthreads();
    }
    if (t == 0) valid[b*4 + oc] = (uint32_t)K;
}

// ---------------- pass 5: compact per batch, emit coords + labels ------------
__global__ void pack(const uint32_t* __restrict__ sel,
                     const uint32_t* __restrict__ valid,
                     int* __restrict__ out)
{
    const int tid = threadIdx.x;        // 256 == B * C*NPTS slots
    if (tid >= B_ * C_ * NPTS) return;
    const int b  = tid >> 7;            // /128
    const int nn = tid & 127;
    uint32_t vc[4] = { valid[b*4+0], valid[b*4+1], valid[b*4+2], valid[b*4+3] };
    int label = -1; uint32_t vox = 0; int have = 0; uint32_t cum = 0;
    #pragma unroll
    for (int oc = 0; oc < 4; ++oc) {
        if (!have && (uint32_t)nn >= cum && (uint32_t)nn < cum + vc[oc]) {
            label = (oc + 1) & 3;
            vox   = sel[(b*4 + oc) * NPTS + (nn - (int)cum)];
            have  = 1;
        }
        cum += vc[oc];
    }
    int x = 0, y = 0, z = 0;
    if (have) { x = (int)(vox >> 14); y = (int)((vox >> 7) & 127); z = (int)(vox & 127); }
    const int base = (b * 128 + nn) * 3;
    out[base + 0] = x; out[base + 1] = y; out[base + 2] = z;
    out[B_ * 128 * 3 + b * 128 + nn] = have ? label : -1;
}

__global__ void init_ws(uint32_t* __restrict__ ws, int n)
{
    int i = blockIdx.x * blockDim.x + threadIdx.x;
    if (i < n) ws[i] = 0;
}

extern "C" void kernel_launch(void* const* d_in, const int* in_sizes, int n_in,
                              void* d_out, int out_size, void* d_ws, size_t ws_size,
                              hipStream_t stream)
{
    (void)in_sizes; (void)n_in; (void)out_size;
    const float* logits = (const float*)d_in[0];
    int*       out  = (int*)d_out;
    uint32_t*  ws   = (uint32_t*)d_ws;
    uint32_t*  hist   = ws + WS_HIST;
    uint32_t*  thresh = ws + WS_THRESH;
    uint32_t*  cnt    = ws + WS_CNT;
    uint32_t*  valid  = ws + WS_VALID;
    uint32_t*  sel    = ws + WS_SEL;

    // candidate capacity from available scratch (deterministic for fixed ws_size)
    long availU32 = (long)(ws_size / 4) - WS_CAND;
    int cap = 8192;
    if (availU32 < (long)2 * B_ * C_ * cap) {
        cap = (int)(availU32 / (2 * B_ * C_));
        if (cap < 64) cap = 64;
    }
    uint32_t* candKey = ws + WS_CAND;
    uint32_t* candIdx = candKey + (size_t)B_ * C_ * cap;

    init_ws<<<(WS_CAND + 255) / 256, 256, 0, stream>>>(ws, WS_CAND);

    dim3 gs(NCHUNK, B_);
    pass1_hist<<<gs, TPB, 0, stream>>>(logits, hist);
    select_thresh<<<1, 32, 0, stream>>>(hist, thresh);
    collect<<<gs, TPB, 0, stream>>>(logits, thresh, cnt, candKey, candIdx, cap);
    topk<<<B_ * C_, TPB, 0, stream>>>(cnt, candKey, candIdx, sel, valid, cap);
    pack<<<1, 256, 0, stream>>>(sel, valid, out);
}